// SSIMLoss_14293651161182
// MI455X (gfx1250) — compile-verified
//
#include <hip/hip_runtime.h>
#include <hip/hip_bf16.h>

typedef __attribute__((ext_vector_type(2))) float v2f;
typedef __attribute__((ext_vector_type(8))) float v8f;

#define IMG_H 512
#define IMG_W 512
#define TILE_R 16      // output rows per workgroup tile
#define TILE_C 64      // output cols per workgroup tile (4 waves x 16)
#define KR 26          // input rows needed: 16 + 11 - 1
#define KRP 28         // K padded to multiple of 4 for wmma 16x16x4
#define HALO_C 74      // 64 + 11 - 1
#define SXW 76         // LDS stride for halo tiles
#define NTAP 11

// ---------------------------------------------------------------------------
// Per-tile SSIM kernel.
//  Stage 1: load zero-padded 26x74 halo of X and Y into LDS.
//  Stage 2: horizontal 11-tap Gaussian on 5 planes (x, y, x2, y2, xy) -> LDS.
//  Stage 3: vertical 11-tap Gaussian as banded-Toeplitz matmul on the matrix
//           pipe: 7 chained v_wmma_f32_16x16x4_f32 per plane per wave.
//  Stage 4: SSIM map + deterministic in-block reduction -> double partial.
// ---------------------------------------------------------------------------
__global__ void __launch_bounds__(128)
ssim_tile_kernel(const float* __restrict__ X, const float* __restrict__ Y,
                 double* __restrict__ partials) {
  __shared__ float gk[NTAP];              // normalized gaussian taps
  __shared__ float Wmat[16][KRP];         // banded Toeplitz weights
  __shared__ float sX[KR][SXW];
  __shared__ float sY[KR][SXW];
  __shared__ float Ht[5][KRP][TILE_C];    // horizontally filtered planes
  __shared__ float red[128];

  const int tid  = threadIdx.x;
  const int lane = tid & 31;
  const int wave = tid >> 5;

  const int bid  = blockIdx.x;
  const int img  = bid >> 8;              // 256 tiles per 512x512 plane
  const int trem = bid & 255;
  const int y0   = (trem >> 3) << 4;      // 32 row tiles
  const int x0   = (trem & 7)  << 6;      // 8 col tiles
  const size_t img_off = (size_t)img * (IMG_H * IMG_W);

  // Gaussian window (WINSIZE=11, SIGMA=1.5), normalized.
  if (tid == 0) {
    float tmp[NTAP];
    float s = 0.f;
    #pragma unroll
    for (int i = 0; i < NTAP; ++i) {
      float d = (float)(i - 5);
      tmp[i] = __expf(-(d * d) / (2.0f * 1.5f * 1.5f));
      s += tmp[i];
    }
    #pragma unroll
    for (int i = 0; i < NTAP; ++i) gk[i] = tmp[i] / s;
  }
  __syncthreads();

  // Banded vertical-conv matrix: W[m][k] = g[k-m] if 0 <= k-m <= 10 else 0.
  for (int i = tid; i < 16 * KRP; i += 128) {
    int m = i / KRP, k = i % KRP;
    int d = k - m;
    Wmat[m][k] = (d >= 0 && d < NTAP) ? gk[d] : 0.f;
  }

  // Load halo tiles (zero padding outside the image, matching 'same' conv).
  for (int i = tid; i < KR * HALO_C; i += 128) {
    int r = i / HALO_C, c = i % HALO_C;
    int gy = y0 - 5 + r, gx = x0 - 5 + c;
    float xv = 0.f, yv = 0.f;
    if (gy >= 0 && gy < IMG_H && gx >= 0 && gx < IMG_W) {
      size_t idx = img_off + (size_t)gy * IMG_W + (size_t)gx;
      xv = X[idx];
      yv = Y[idx];
    }
    sX[r][c] = xv;
    sY[r][c] = yv;
  }
  __syncthreads();

  // Horizontal pass on all 5 planes at once (products formed in registers).
  for (int i = tid; i < KR * TILE_C; i += 128) {
    int r = i / TILE_C, c = i % TILE_C;
    float hx = 0.f, hy = 0.f, hxx = 0.f, hyy = 0.f, hxy = 0.f;
    #pragma unroll
    for (int j = 0; j < NTAP; ++j) {
      float g  = gk[j];
      float xs = sX[r][c + j];
      float ys = sY[r][c + j];
      hx  = fmaf(g, xs,      hx);
      hy  = fmaf(g, ys,      hy);
      hxx = fmaf(g, xs * xs, hxx);
      hyy = fmaf(g, ys * ys, hyy);
      hxy = fmaf(g, xs * ys, hxy);
    }
    Ht[0][r][c] = hx;  Ht[1][r][c] = hy;
    Ht[2][r][c] = hxx; Ht[3][r][c] = hyy; Ht[4][r][c] = hxy;
  }
  // K-padding rows must be clean (weights there are 0, but 0*NaN = NaN).
  if (tid < TILE_C) {
    #pragma unroll
    for (int p = 0; p < 5; ++p) {
      Ht[p][26][tid] = 0.f;
      Ht[p][27][tid] = 0.f;
    }
  }
  __syncthreads();

  // --- Vertical pass: D(16x16) += A(16x4) x B(4x16), 7 k-steps, 5 planes ---
  // A layout (f32 16x4): lanes 0-15 hold M=lane, K={0,1} in v0/v1;
  //                      lanes 16-31 hold M=lane-16, K={2,3}.
  const int m     = lane & 15;
  const int kbase = (lane < 16) ? 0 : 2;
  const int ncol  = (wave << 4) + m;      // this wave's 16-col sub-tile

  v2f aw[7];
  #pragma unroll
  for (int t = 0; t < 7; ++t) {
    aw[t].x = Wmat[m][4 * t + kbase];
    aw[t].y = Wmat[m][4 * t + kbase + 1];
  }

  v8f dacc[5];
  #pragma unroll
  for (int p = 0; p < 5; ++p) {
    v8f cacc = {0.f, 0.f, 0.f, 0.f, 0.f, 0.f, 0.f, 0.f};
    #pragma unroll
    for (int t = 0; t < 7; ++t) {
      v2f b;
      b.x = Ht[p][4 * t + kbase][ncol];
      b.y = Ht[p][4 * t + kbase + 1][ncol];
      // (neg_a, A, neg_b, B, c_mod, C, reuse_a, reuse_b)
      cacc = __builtin_amdgcn_wmma_f32_16x16x4_f32(
          false, aw[t], false, b, (short)0, cacc, false, false);
    }
    dacc[p] = cacc;
  }

  // SSIM map + per-lane partial sum over this lane's 8 output elements.
  const float C1 = 1e-4f, C2 = 9e-4f;
  float lsum = 0.f;
  #pragma unroll
  for (int v = 0; v < 8; ++v) {
    float mu1 = dacc[0][v], mu2 = dacc[1][v];
    float ex2 = dacc[2][v], ey2 = dacc[3][v], exy = dacc[4][v];
    float mu1s = mu1 * mu1, mu2s = mu2 * mu2, mu12 = mu1 * mu2;
    float s1  = ex2 - mu1s;
    float s2  = ey2 - mu2s;
    float s12 = exy - mu12;
    float num = (2.f * mu12 + C1) * (2.f * s12 + C2);
    float den = (mu1s + mu2s + C1) * (s1 + s2 + C2);
    lsum += num / den;
  }

  // Deterministic block reduction (fixed order), one double partial per block.
  red[tid] = lsum;
  __syncthreads();
  if (tid == 0) {
    double s = 0.0;
    for (int i = 0; i < 128; ++i) s += (double)red[i];
    partials[bid] = s;
  }
}

// Deterministic final reduction of all block partials -> 1 - mean(ssim_map).
__global__ void __launch_bounds__(256)
ssim_reduce_kernel(const double* __restrict__ partials, int nparts,
                   float* __restrict__ out, double inv_total) {
  __shared__ double red[256];
  const int tid = threadIdx.x;
  double s = 0.0;
  for (int i = tid; i < nparts; i += 256) s += partials[i];
  red[tid] = s;
  __syncthreads();
  #pragma unroll
  for (int off = 128; off > 0; off >>= 1) {
    if (tid < off) red[tid] += red[tid + off];
    __syncthreads();
  }
  if (tid == 0) out[0] = (float)(1.0 - red[0] * inv_total);
}

extern "C" void kernel_launch(void* const* d_in, const int* in_sizes, int n_in,
                              void* d_out, int out_size, void* d_ws, size_t ws_size,
                              hipStream_t stream) {
  (void)in_sizes; (void)n_in; (void)out_size; (void)ws_size;
  const float* X = (const float*)d_in[0];
  const float* Y = (const float*)d_in[1];
  float* out = (float*)d_out;
  double* partials = (double*)d_ws;   // 12288 doubles = 96 KB of scratch

  const int nblocks = 48 * 32 * 8;    // 48 planes x (32 row-tiles x 8 col-tiles)
  ssim_tile_kernel<<<nblocks, 128, 0, stream>>>(X, Y, partials);

  const double inv_total = 1.0 / (16.0 * 3.0 * 512.0 * 512.0);
  ssim_reduce_kernel<<<1, 256, 0, stream>>>(partials, nblocks, out, inv_total);
}